// LRUBlock_5729486373073
// MI455X (gfx1250) — compile-verified
//
#include <hip/hip_runtime.h>
#include <hip/hip_bf16.h>

// ---------------------------------------------------------------------------
// LRU block for MI455X (gfx1250), wave32, FP32 WMMA 16x16x4.
//
// Shapes: B=8, L=4096, IN=256, DM=512, OUT=256, M = B*L = 32768
//   u   = inputs @ (Bc * e^gamma)^T        (complex, via 2 real f32 WMMA GEMMs,
//                                           gamma folded into the epilogue)
//   x[t]= A*x[t-1] + u[t]                  (chunked parallel scan, exact)
//   y   = x_flat @ W2^T + inputs @ D^T     (W2 interleaves C_re / -C_im so the
//                                           complex->real GEMM is one real GEMM)
// ---------------------------------------------------------------------------

typedef __attribute__((ext_vector_type(2))) float v2f;
typedef __attribute__((ext_vector_type(8))) float v8f;

#define M_TOT   32768
#define L_SEQ   4096
#define IN_DIM  256
#define D_MODEL 512
#define OUT_DIM 256
#define K_Y     (2 * D_MODEL)   // 1024: interleaved (re,im) as a real K dim
#define NCHUNK  64
#define CHUNK_L 64              // L_SEQ / NCHUNK

// ---------------------------------------------------------------------------
// u = inputs @ (Bc * e^gamma)^T   -> float2 [M, D_MODEL] interleaved (re, im)
// One wave = one 16x16 tile, two accumulators (re / im).
// block = 8 waves arranged 4(M) x 2(N); grid = (D_MODEL/32, M/64)
// ---------------------------------------------------------------------------
__global__ void __launch_bounds__(256)
lru_proj_u(const float* __restrict__ in,     // [M, IN_DIM]
           const float* __restrict__ Bre,    // [D_MODEL, IN_DIM]
           const float* __restrict__ Bim,    // [D_MODEL, IN_DIM]
           const float* __restrict__ gamma_, // [D_MODEL]
           float2* __restrict__ u)           // [M, D_MODEL]
{
    const int lane = threadIdx.x & 31;
    const int wave = threadIdx.x >> 5;
    const int wm = wave & 3, wn = wave >> 2;
    const int M0 = (blockIdx.y * 4 + wm) * 16;
    const int N0 = (blockIdx.x * 2 + wn) * 16;
    const int ml = lane & 15;      // M (for A) / N (for B) within tile
    const int hh = lane >> 4;      // selects K pair {0,1} vs {2,3}

    const int n = N0 + ml;
    const float g = __expf(gamma_[n]);   // applied in epilogue (GEMM linear in B)

    const float* aPtr  = in  + (size_t)(M0 + ml) * IN_DIM + 2 * hh;
    const float* brPtr = Bre + (size_t)n * IN_DIM + 2 * hh;
    const float* biPtr = Bim + (size_t)n * IN_DIM + 2 * hh;

    v8f accR = {}; v8f accI = {};
    #pragma unroll 4
    for (int k = 0; k < IN_DIM; k += 4) {
        v2f a  = *(const v2f*)(aPtr  + k);
        v2f br = *(const v2f*)(brPtr + k);
        v2f bi = *(const v2f*)(biPtr + k);
        accR = __builtin_amdgcn_wmma_f32_16x16x4_f32(false, a, false, br,
                                                     (short)0, accR, false, false);
        accI = __builtin_amdgcn_wmma_f32_16x16x4_f32(false, a, false, bi,
                                                     (short)0, accI, false, false);
    }
    #pragma unroll
    for (int r = 0; r < 8; ++r) {
        size_t idx = (size_t)(M0 + r + 8 * hh) * D_MODEL + n;
        u[idx] = make_float2(accR[r] * g, accI[r] * g);
    }
}

// ---------------------------------------------------------------------------
// Scan helpers
// ---------------------------------------------------------------------------
__device__ __forceinline__ float2 lru_A(const float* nu_log, const float* th_log, int d) {
    float mag = __expf(-__expf(nu_log[d]));
    float th  = __expf(th_log[d]);
    return make_float2(mag * __cosf(th), mag * __sinf(th));
}
__device__ __forceinline__ float2 cmul(float2 a, float2 b) {
    return make_float2(a.x * b.x - a.y * b.y, a.x * b.y + a.y * b.x);
}
__device__ __forceinline__ float2 cstep(float2 A, float2 x, float2 uv) {
    // A*x + u
    return make_float2(fmaf(A.x, x.x, fmaf(-A.y, x.y, uv.x)),
                       fmaf(A.x, x.y, fmaf( A.y, x.x, uv.y)));
}

// Pass A: local chunk scan (zero init), keep only end state.
// grid = B*NCHUNK blocks, block = 512 threads (one per channel d)
__global__ void __launch_bounds__(512)
lru_scan_partial(const float2* __restrict__ u,
                 const float* __restrict__ nu_log,
                 const float* __restrict__ th_log,
                 float2* __restrict__ endBuf)   // [B*NCHUNK, D_MODEL]
{
    const int d = threadIdx.x;
    const float2 A = lru_A(nu_log, th_log, d);
    const float2* up = u + (size_t)blockIdx.x * CHUNK_L * D_MODEL + d;
    float2 xv = make_float2(0.f, 0.f);
    #pragma unroll 8
    for (int j = 0; j < CHUNK_L; ++j)
        xv = cstep(A, xv, up[(size_t)j * D_MODEL]);
    endBuf[(size_t)blockIdx.x * D_MODEL + d] = xv;
}

// Pass B: serial scan over chunk carries per batch with Ac = A^CHUNK_L.
// grid = B blocks, block = 512 threads
__global__ void __launch_bounds__(512)
lru_scan_carry(const float2* __restrict__ endBuf,
               const float* __restrict__ nu_log,
               const float* __restrict__ th_log,
               float2* __restrict__ carryBuf)  // [B*NCHUNK, D_MODEL]
{
    const int d = threadIdx.x;
    const float2 A = lru_A(nu_log, th_log, d);
    float2 Ac = A;                      // A^(2^6) = A^64 = A^CHUNK_L
    #pragma unroll
    for (int i = 0; i < 6; ++i) Ac = cmul(Ac, Ac);
    float2 c = make_float2(0.f, 0.f);
    for (int ci = 0; ci < NCHUNK; ++ci) {
        size_t idx = ((size_t)blockIdx.x * NCHUNK + ci) * D_MODEL + d;
        carryBuf[idx] = c;              // state entering chunk ci
        c = cstep(Ac, c, endBuf[idx]);
    }
}

// Pass C: re-run each chunk seeded with its carry, write x (complex64 layout).
// grid = B*NCHUNK blocks, block = 512 threads
__global__ void __launch_bounds__(512)
lru_scan_final(const float2* __restrict__ u,
               const float* __restrict__ nu_log,
               const float* __restrict__ th_log,
               const float2* __restrict__ carryBuf,
               float2* __restrict__ xOut)      // [M, D_MODEL]
{
    const int d = threadIdx.x;
    const float2 A = lru_A(nu_log, th_log, d);
    float2 xv = carryBuf[(size_t)blockIdx.x * D_MODEL + d];
    const float2* up = u    + (size_t)blockIdx.x * CHUNK_L * D_MODEL + d;
    float2*       xp = xOut + (size_t)blockIdx.x * CHUNK_L * D_MODEL + d;
    #pragma unroll 8
    for (int j = 0; j < CHUNK_L; ++j) {
        xv = cstep(A, xv, up[(size_t)j * D_MODEL]);
        xp[(size_t)j * D_MODEL] = xv;
    }
}

// ---------------------------------------------------------------------------
// W2[n][2d]   =  C_re[n][d]
// W2[n][2d+1] = -C_im[n][d]
// so that  Re(x @ Cc^T)[m][n] = sum_k x_flat[m][k] * W2[n][k],  k in [0,1024)
// grid = OUT_DIM blocks, block = 512 threads
// ---------------------------------------------------------------------------
__global__ void __launch_bounds__(512)
lru_prep_w2(const float* __restrict__ Cre,   // [OUT_DIM, D_MODEL]
            const float* __restrict__ Cim,   // [OUT_DIM, D_MODEL]
            float2* __restrict__ W2)         // [OUT_DIM, D_MODEL] of (re,-im)
{
    const int n = blockIdx.x;
    const int d = threadIdx.x;
    W2[(size_t)n * D_MODEL + d] =
        make_float2(Cre[(size_t)n * D_MODEL + d], -Cim[(size_t)n * D_MODEL + d]);
}

// ---------------------------------------------------------------------------
// y = x_flat @ W2^T + inputs @ D^T
// x viewed as real [M, 1024]; single accumulator, one WMMA stream, no shuffles.
// block = 8 waves 4(M) x 2(N); grid = (OUT_DIM/32, M/64)
// ---------------------------------------------------------------------------
__global__ void __launch_bounds__(256)
lru_proj_y(const float* __restrict__ xf,    // [M, K_Y]  (x interleaved re/im)
           const float* __restrict__ W2,    // [OUT_DIM, K_Y]
           const float* __restrict__ in,    // [M, IN_DIM]
           const float* __restrict__ Dw,    // [OUT_DIM, IN_DIM]
           float* __restrict__ y)           // [M, OUT_DIM]
{
    const int lane = threadIdx.x & 31;
    const int wave = threadIdx.x >> 5;
    const int wm = wave & 3, wn = wave >> 2;
    const int M0 = (blockIdx.y * 4 + wm) * 16;
    const int N0 = (blockIdx.x * 2 + wn) * 16;
    const int ml = lane & 15;
    const int hh = lane >> 4;

    const int m = M0 + ml;
    const int n = N0 + ml;

    const float* aPtr = xf + (size_t)m * K_Y + 2 * hh;
    const float* bPtr = W2 + (size_t)n * K_Y + 2 * hh;

    v8f acc = {};
    #pragma unroll 4
    for (int k = 0; k < K_Y; k += 4) {
        v2f a = *(const v2f*)(aPtr + k);
        v2f b = *(const v2f*)(bPtr + k);
        acc = __builtin_amdgcn_wmma_f32_16x16x4_f32(false, a, false, b,
                                                    (short)0, acc, false, false);
    }

    // fused skip: + inputs @ D^T
    const float* sPtr = in + (size_t)m * IN_DIM + 2 * hh;
    const float* dPtr = Dw + (size_t)n * IN_DIM + 2 * hh;
    #pragma unroll 4
    for (int k = 0; k < IN_DIM; k += 4) {
        v2f a = *(const v2f*)(sPtr + k);
        v2f b = *(const v2f*)(dPtr + k);
        acc = __builtin_amdgcn_wmma_f32_16x16x4_f32(false, a, false, b,
                                                    (short)0, acc, false, false);
    }

    #pragma unroll
    for (int r = 0; r < 8; ++r)
        y[(size_t)(M0 + r + 8 * hh) * OUT_DIM + n] = acc[r];
}

// ---------------------------------------------------------------------------
extern "C" void kernel_launch(void* const* d_in, const int* in_sizes, int n_in,
                              void* d_out, int out_size, void* d_ws, size_t ws_size,
                              hipStream_t stream) {
    const float* inputs    = (const float*)d_in[0];  // [8,4096,256]
    const float* nu_log    = (const float*)d_in[1];  // [512]
    const float* theta_log = (const float*)d_in[2];  // [512]
    const float* gamma_log = (const float*)d_in[3];  // [512]
    const float* B_re      = (const float*)d_in[4];  // [512,256]
    const float* B_im      = (const float*)d_in[5];  // [512,256]
    const float* C_re      = (const float*)d_in[6];  // [256,512]
    const float* C_im      = (const float*)d_in[7];  // [256,512]
    const float* Dw        = (const float*)d_in[8];  // [256,256]

    // d_out = [ x : complex64 [8,4096,512] interleaved | y : f32 [8,4096,256] ]
    float2* xOut = (float2*)d_out;
    float*  yOut = (float*)d_out + (size_t)M_TOT * D_MODEL * 2;

    // workspace layout
    char* ws = (char*)d_ws;
    float2* u        = (float2*)ws;                                   // 128 MB
    float2* endBuf   = (float2*)(ws + (size_t)M_TOT * D_MODEL * 8);   //   2 MB
    float2* carryBuf = endBuf + (size_t)8 * NCHUNK * D_MODEL;         //   2 MB
    float2* W2       = carryBuf + (size_t)8 * NCHUNK * D_MODEL;       //   1 MB

    // 0) build W2 = interleaved (C_re, -C_im)
    lru_prep_w2<<<dim3(OUT_DIM), 512, 0, stream>>>(C_re, C_im, W2);

    // 1) u projection (complex GEMM, f32 WMMA)
    lru_proj_u<<<dim3(D_MODEL / 32, M_TOT / 64), 256, 0, stream>>>(
        inputs, B_re, B_im, gamma_log, u);

    // 2) chunked linear recurrence x[t] = A x[t-1] + u[t]
    lru_scan_partial<<<dim3(8 * NCHUNK), 512, 0, stream>>>(u, nu_log, theta_log, endBuf);
    lru_scan_carry  <<<dim3(8),          512, 0, stream>>>(endBuf, nu_log, theta_log, carryBuf);
    lru_scan_final  <<<dim3(8 * NCHUNK), 512, 0, stream>>>(u, nu_log, theta_log, carryBuf, xOut);

    // 3) y projection (one real GEMM over K=1024 + fused skip, f32 WMMA)
    lru_proj_y<<<dim3(OUT_DIM / 32, M_TOT / 64), 256, 0, stream>>>(
        (const float*)xOut, (const float*)W2, inputs, Dw, yOut);
}